// DGM_2362232013242
// MI455X (gfx1250) — compile-verified
//
#include <hip/hip_runtime.h>
#include <cmath>

typedef __attribute__((__ext_vector_type__(2))) float v2f;
typedef __attribute__((__ext_vector_type__(8))) float v8f;

#define HWXY 4096   // 64*64
#define CIN  128
#define CCH  128    // channel chunk per gather workgroup

// ---------------- Kernel 1: fm = mean over 128 channels -------------------
__global__ __launch_bounds__(256) void mean_kernel(const float* __restrict__ feat,
                                                   float* __restrict__ fm) {
    int i = blockIdx.x * 256 + threadIdx.x;      // 0..8191 = b*4096 + p*64 + q
    int b = i >> 12;
    int s = i & 4095;
    const float* fp = feat + (size_t)b * CIN * HWXY + s;
    float acc = 0.0f;
#pragma unroll 8
    for (int c = 0; c < CIN; ++c) acc += fp[(size_t)c * HWXY];
    fm[i] = acc * (1.0f / 128.0f);
}

// ---------------- Kernel 2: offset head via V_WMMA_F32_16X16X4_F32 --------
// One wave (32 threads) handles 16 positions. A = z (16 pos x 32 ch, chunked
// K=4), B columns 0/1 = W_ofs rows, D[:,0]=oy, D[:,1]=ox. Then per-position
// scalar math -> tap offsets (int4) + ga-premultiplied bilinear weights
// (float4) into workspace.
__global__ __launch_bounds__(32) void params_kernel(
    const float* __restrict__ fm,
    const float* __restrict__ W_in,  const float* __restrict__ b_in,
    const float* __restrict__ W_dw,  const float* __restrict__ b_dw,
    const float* __restrict__ W_ofs, const float* __restrict__ W_a,
    const float* __restrict__ b_a,   const float* __restrict__ lam,
    int4* __restrict__ poffs, float4* __restrict__ pwts)
{
    const int lane = threadIdx.x;
    const int half = lane >> 4;          // K-half select (f32 16x16x4 layout)
    const int m    = lane & 15;          // A row / B column index
    const int pos  = blockIdx.x * 16 + m;
    const int b = pos >> 12, p = (pos >> 6) & 63, q = pos & 63;

    // 5x5 replicate-padded neighborhood of fm
    const float* fmb = fm + (b << 12);
    float nb[25];
#pragma unroll
    for (int u = 0; u < 5; ++u) {
        int pp = p + u - 2; pp = pp < 0 ? 0 : (pp > 63 ? 63 : pp);
#pragma unroll
        for (int v = 0; v < 5; ++v) {
            int qq = q + v - 2; qq = qq < 0 ? 0 : (qq > 63 ? 63 : qq);
            nb[u * 5 + v] = fmb[pp * 64 + qq];
        }
    }

    // z[c] = relu(W_in[c]*conv5x5(fm) + b_in[c]*sum(W_dw[c]) + b_dw[c])
    // this lane supplies A elements (m, k) for k = 4*kc + 2*half + {0,1}
    float zb[16];
#pragma unroll
    for (int kc = 0; kc < 8; ++kc) {
#pragma unroll
        for (int s = 0; s < 2; ++s) {
            int c = 4 * kc + 2 * half + s;
            float acc = 0.0f, sw = 0.0f;
#pragma unroll
            for (int j = 0; j < 25; ++j) {
                float wv = W_dw[c * 25 + j];
                acc += wv * nb[j];
                sw  += wv;
            }
            float zv = W_in[c] * acc + b_in[c] * sw + b_dw[c];
            zb[2 * kc + s] = zv > 0.0f ? zv : 0.0f;
        }
    }

    // B fragment: column n==0 -> W_ofs row 0 (y), n==1 -> W_ofs row 1 (x)
    float bx[8], by[8];
#pragma unroll
    for (int kc = 0; kc < 8; ++kc) {
        int k0 = 4 * kc + 2 * half;
        bx[kc] = (m == 0) ? W_ofs[k0]       : ((m == 1) ? W_ofs[32 + k0]     : 0.0f);
        by[kc] = (m == 0) ? W_ofs[k0 + 1]   : ((m == 1) ? W_ofs[32 + k0 + 1] : 0.0f);
    }

    v8f acc = {};
#pragma unroll
    for (int kc = 0; kc < 8; ++kc) {
        v2f A;  A.x  = zb[2 * kc]; A.y  = zb[2 * kc + 1];
        v2f Bf; Bf.x = bx[kc];     Bf.y = by[kc];
        acc = __builtin_amdgcn_wmma_f32_16x16x4_f32(
            false, A, false, Bf, (short)0, acc, false, false);
    }

    // Unpack D through LDS: VGPR g holds D[M = g + 8*half][N = m]
    __shared__ float D[16 * 16];
#pragma unroll
    for (int g = 0; g < 8; ++g) D[(g + 8 * half) * 16 + m] = acc[g];
    __syncthreads();

    if (lane < 16) {
        int posL = blockIdx.x * 16 + lane;
        int pp = (posL >> 6) & 63, qq = posL & 63;
        float oy = D[lane * 16 + 0];
        float ox = D[lane * 16 + 1];
        float osy = tanhf(oy) * (1.0f / 64.0f);
        float osx = tanhf(ox) * (1.0f / 64.0f);
        float ry = ((float)pp + 0.5f) * (2.0f / 64.0f) - 1.0f;
        float rx = ((float)qq + 0.5f) * (2.0f / 64.0f) - 1.0f;
        float py = fminf(fmaxf(100.0f * osy + ry, -1.0f), 1.0f);
        float px = fminf(fmaxf(100.0f * osx + rx, -1.0f), 1.0f);
        float x = (px + 1.0f) * 0.5f * 63.0f;
        float y = (py + 1.0f) * 0.5f * 63.0f;
        float x0 = floorf(x), y0 = floorf(y);
        float wx1 = x - x0, wx0 = 1.0f - wx1;
        float wy1 = y - y0, wy0 = 1.0f - wy1;
        int x0i = (int)fminf(fmaxf(x0,        0.0f), 63.0f);
        int x1i = (int)fminf(fmaxf(x0 + 1.0f, 0.0f), 63.0f);
        int y0i = (int)fminf(fmaxf(y0,        0.0f), 63.0f);
        int y1i = (int)fminf(fmaxf(y0 + 1.0f, 0.0f), 63.0f);

        // ga = 1 + lam*(sum(W_a*W_in)*fm + sum(W_a*b_in) + b_a)
        float dWaWin = 0.0f, dWaBin = 0.0f;
#pragma unroll 4
        for (int c = 0; c < 32; ++c) { dWaWin += W_a[c] * W_in[c]; dWaBin += W_a[c] * b_in[c]; }
        float ga = 1.0f + lam[0] * (dWaWin * fm[posL] + dWaBin + b_a[0]);

        int4 of;
        of.x = (y0i * 64 + x0i) << 12;
        of.y = (y0i * 64 + x1i) << 12;
        of.z = (y1i * 64 + x0i) << 12;
        of.w = (y1i * 64 + x1i) << 12;
        float4 wt = make_float4(wy0 * wx0 * ga, wy0 * wx1 * ga,
                                wy1 * wx0 * ga, wy1 * wx1 * ga);
        poffs[posL] = of;
        pwts[posL]  = wt;
    }
}

// ---------------- Kernel 3: bandwidth-bound gather + LDS transpose --------
// Workgroup = (b, p, 128-channel chunk). Phase A: wave w gathers q = qi*8+w,
// 4 taps x 128 contiguous channels via float4 (coalesced reads), blends with
// premultiplied weights, writes into padded LDS tile T[c][q]. Phase B: read
// transposed, store 64-float contiguous rows of out (coalesced writes).
__global__ __launch_bounds__(256) void gather_kernel(
    const float* __restrict__ gauss,
    const int4* __restrict__ poffs, const float4* __restrict__ pwts,
    float* __restrict__ out)
{
    __shared__ float  T[CCH * 65];
    __shared__ int4   soff[64];
    __shared__ float4 swt[64];
    const int t     = threadIdx.x;
    const int chunk = blockIdx.x;   // 0..31
    const int p     = blockIdx.y;   // 0..63
    const int b     = blockIdx.z;   // 0..1
    const int cbase = chunk * CCH;

    if (t < 64) {
        int pi = ((b << 6) + p) * 64 + t;   // b*4096 + p*64 + q
        soff[t] = poffs[pi];
        swt[t]  = pwts[pi];
    }
    __syncthreads();

    const float* gb  = gauss + (size_t)b * HWXY * HWXY;
    const int wave = t >> 5, lane = t & 31;
#pragma unroll 2
    for (int qi = 0; qi < 8; ++qi) {
        int q = qi * 8 + wave;
        int4   of = soff[q];
        float4 wt = swt[q];
        int c  = cbase + lane * 4;
        const float4 v00 = *(const float4*)(gb + of.x + c);
        const float4 v01 = *(const float4*)(gb + of.y + c);
        const float4 v10 = *(const float4*)(gb + of.z + c);
        const float4 v11 = *(const float4*)(gb + of.w + c);
        int cl = lane * 4;
        T[(cl + 0) * 65 + q] = wt.x * v00.x + wt.y * v01.x + wt.z * v10.x + wt.w * v11.x;
        T[(cl + 1) * 65 + q] = wt.x * v00.y + wt.y * v01.y + wt.z * v10.y + wt.w * v11.y;
        T[(cl + 2) * 65 + q] = wt.x * v00.z + wt.y * v01.z + wt.z * v10.z + wt.w * v11.z;
        T[(cl + 3) * 65 + q] = wt.x * v00.w + wt.y * v01.w + wt.z * v10.w + wt.w * v11.w;
    }
    __syncthreads();

    const int q  = t & 63;
    const int cr = t >> 6;
    size_t outrow = (size_t)b * HWXY * HWXY + (size_t)p * 64 + q;
#pragma unroll 4
    for (int c = cr; c < CCH; c += 4) {
        out[outrow + (size_t)(cbase + c) * HWXY] = T[c * 65 + q];
    }
}

// ---------------------------------------------------------------------------
extern "C" void kernel_launch(void* const* d_in, const int* in_sizes, int n_in,
                              void* d_out, int out_size, void* d_ws, size_t ws_size,
                              hipStream_t stream) {
    const float* feat  = (const float*)d_in[0];
    const float* gauss = (const float*)d_in[1];
    const float* W_in  = (const float*)d_in[2];
    const float* b_in  = (const float*)d_in[3];
    const float* W_dw  = (const float*)d_in[4];
    const float* b_dw  = (const float*)d_in[5];
    const float* W_ofs = (const float*)d_in[6];
    const float* W_a   = (const float*)d_in[7];
    const float* b_a   = (const float*)d_in[8];
    const float* lam   = (const float*)d_in[9];
    float* out = (float*)d_out;

    // workspace: fm (8192 f32) | poffs (8192 int4) | pwts (8192 float4)
    float* fm    = (float*)d_ws;
    int4*  poffs = (int4*)((char*)d_ws + 32768);
    float4* pwts = (float4*)((char*)d_ws + 32768 + 131072);

    mean_kernel<<<32, 256, 0, stream>>>(feat, fm);
    params_kernel<<<512, 32, 0, stream>>>(fm, W_in, b_in, W_dw, b_dw,
                                          W_ofs, W_a, b_a, lam, poffs, pwts);
    gather_kernel<<<dim3(32, 64, 2), 256, 0, stream>>>(gauss, poffs, pwts, out);
}